// NCC_43568148251076
// MI455X (gfx1250) — compile-verified
//
#include <hip/hip_runtime.h>

// ---------------------------------------------------------------------------
// Local NCC loss, win=9, on [2,1,192,192,192] f32.
// Separable 9-tap box filter expressed as banded matmul on V_WMMA_F32_16X16X4_F32.
// ---------------------------------------------------------------------------

typedef float v2f __attribute__((ext_vector_type(2)));
typedef float v8f __attribute__((ext_vector_type(8)));

constexpr int W = 192, H = 192, D = 192, B = 2;
constexpr int SLICE = H * W;                 // 36864
constexpr int NVOX  = B * D * H * W;         // 14,155,776
constexpr int TIL   = 16;
constexpr int STEPS = 6;                     // 24 ext positions / K=4
constexpr int TPL   = W / TIL;               // 12
constexpr int TILES = B * D * TPL * TPL;     // 55,296 (same for every pass)
constexpr int WAVES = 8;                     // waves per 256-thread block
constexpr int BLOCKS = TILES / WAVES;        // 6,912
constexpr float INV_WIN3 = 1.0f / 729.0f;
constexpr float EPS = 1e-5f;

__device__ __forceinline__ v2f mk2(float a, float b) { v2f v; v[0] = a; v[1] = b; return v; }

// D = A(16x4) * B(4x16) + C, all f32, wave32.
__device__ __forceinline__ v8f wmma4(v2f a, v2f b, v8f c) {
    return __builtin_amdgcn_wmma_f32_16x16x4_f32(false, a, false, b, (short)0, c,
                                                 false, false);
}

// Band matrix S[m,j] = 1 iff j-m in [0,8]  (9-tap box, pad=4).
__device__ __forceinline__ float bandA(int j, int m) {
    return (j >= m && j <= m + 8) ? 1.0f : 0.0f;
}

// ---------------------------------------------------------------------------
// Pass 1: elementwise products + box-sum along W (contiguous axis), via LDS.
// Writes 5 volumes: I, J, I2, J2, IJ  (field stride NVOX).
// ---------------------------------------------------------------------------
__global__ __launch_bounds__(256) void ncc_pass1(const float* __restrict__ yp,
                                                 const float* __restrict__ yt,
                                                 float* __restrict__ outA) {
    __shared__ float sP[WAVES][TIL][25];   // 25: bank-conflict-free stride
    __shared__ float sT[WAVES][TIL][25];

    const int lane = threadIdx.x & 31;
    const int wave = threadIdx.x >> 5;
    const int tile = blockIdx.x * WAVES + wave;
    const int slice = tile / (TPL * TPL);        // b*D + d
    const int rem   = tile % (TPL * TPL);
    const int h0 = (rem / TPL) * TIL;
    const int w0 = (rem % TPL) * TIL;
    const int base = slice * SLICE;

    // Stage 16(h) x 24(w-ext) patches with zero padding at W edges.
#pragma unroll
    for (int i = 0; i < 12; ++i) {
        int idx = i * 32 + lane;
        int r = idx / 24, c = idx % 24;
        int gw = w0 - 4 + c;
        bool ok = (gw >= 0) && (gw < W);
        int a = base + (h0 + r) * W + (ok ? gw : 0);
        sP[wave][r][c] = ok ? yp[a] : 0.0f;
        sT[wave][r][c] = ok ? yt[a] : 0.0f;
    }
    __syncthreads();

    const int n    = lane & 15;   // B/D column (h-line); also A's row m (w out pos)
    const int half = lane >> 4;

    v8f aI = {}, aJ = {}, aI2 = {}, aJ2 = {}, aIJ = {};
#pragma unroll
    for (int s = 0; s < STEPS; ++s) {
        int j0 = 4 * s + 2 * half;
        float p0 = sP[wave][n][j0], p1 = sP[wave][n][j0 + 1];
        float t0 = sT[wave][n][j0], t1 = sT[wave][n][j0 + 1];
        v2f A = mk2(bandA(j0, n), bandA(j0 + 1, n));
        aI  = wmma4(A, mk2(p0, p1), aI);
        aJ  = wmma4(A, mk2(t0, t1), aJ);
        aI2 = wmma4(A, mk2(p0 * p0, p1 * p1), aI2);
        aJ2 = wmma4(A, mk2(t0 * t0, t1 * t1), aJ2);
        aIJ = wmma4(A, mk2(p0 * t0, p1 * t1), aIJ);
    }

    // D layout: lane holds h-line n; M (w position) = g + 8*half, contiguous.
    const int ob = base + (h0 + n) * W + w0 + 8 * half;
#define NCC_ST(acc, f)                                                              \
    {                                                                               \
        float* o = outA + (size_t)(f) * NVOX + ob;                                  \
        *(float4*)(o)     = make_float4(acc[0], acc[1], acc[2], acc[3]);            \
        *(float4*)(o + 4) = make_float4(acc[4], acc[5], acc[6], acc[7]);            \
    }
    NCC_ST(aI, 0) NCC_ST(aJ, 1) NCC_ST(aI2, 2) NCC_ST(aJ2, 3) NCC_ST(aIJ, 4)
#undef NCC_ST
}

// ---------------------------------------------------------------------------
// Pass 2: box-sum along H for all 5 fields. Lines run along W -> coalesced.
// ---------------------------------------------------------------------------
__global__ __launch_bounds__(256) void ncc_pass2(const float* __restrict__ src,
                                                 float* __restrict__ dst) {
    const int lane = threadIdx.x & 31;
    const int wave = threadIdx.x >> 5;
    const int tile = blockIdx.x * WAVES + wave;
    const int slice = tile / (TPL * TPL);        // b*D + d
    const int rem   = tile % (TPL * TPL);
    const int h0 = (rem / TPL) * TIL;            // filtered-axis output base
    const int w0 = (rem % TPL) * TIL;            // line base
    const int base = slice * SLICE;
    const int n    = lane & 15;
    const int half = lane >> 4;

#pragma unroll
    for (int f = 0; f < 5; ++f) {
        const float* s = src + (size_t)f * NVOX;
        v8f acc = {};
#pragma unroll
        for (int st = 0; st < STEPS; ++st) {
            int j0 = 4 * st + 2 * half;
            int r0 = h0 - 4 + j0, r1 = r0 + 1;
            bool ok0 = (r0 >= 0) && (r0 < H);
            bool ok1 = (r1 >= 0) && (r1 < H);
            float b0 = s[base + (ok0 ? r0 : 0) * W + w0 + n];
            float b1 = s[base + (ok1 ? r1 : 0) * W + w0 + n];
            b0 = ok0 ? b0 : 0.0f;
            b1 = ok1 ? b1 : 0.0f;
            v2f A = mk2(bandA(j0, n), bandA(j0 + 1, n));
            acc = wmma4(A, mk2(b0, b1), acc);
        }
        float* o = dst + (size_t)f * NVOX;
#pragma unroll
        for (int g = 0; g < 8; ++g) {
            int m = g + 8 * half;                 // h output offset
            o[base + (h0 + m) * W + w0 + n] = acc[g];
        }
    }
}

// ---------------------------------------------------------------------------
// Pass 3: box-sum along D (per sample b), fused NCC combine + block reduction.
// ---------------------------------------------------------------------------
__global__ __launch_bounds__(256) void ncc_pass3(const float* __restrict__ src,
                                                 float* __restrict__ partial) {
    __shared__ float red[256];
    const int lane = threadIdx.x & 31;
    const int wave = threadIdx.x >> 5;
    const int tile = blockIdx.x * WAVES + wave;
    const int sl  = tile / (TPL * TPL);          // b*H + h
    const int rem = tile % (TPL * TPL);
    const int b  = sl / H;
    const int h  = sl % H;
    const int d0 = (rem / TPL) * TIL;
    const int w0 = (rem % TPL) * TIL;
    const int n    = lane & 15;
    const int half = lane >> 4;

    v8f acc[5];
#pragma unroll
    for (int f = 0; f < 5; ++f) {
        const float* s = src + (size_t)f * NVOX;
        v8f a = {};
#pragma unroll
        for (int st = 0; st < STEPS; ++st) {
            int j0 = 4 * st + 2 * half;
            int r0 = d0 - 4 + j0, r1 = r0 + 1;
            bool ok0 = (r0 >= 0) && (r0 < D);
            bool ok1 = (r1 >= 0) && (r1 < D);
            float b0 = s[(b * D + (ok0 ? r0 : 0)) * SLICE + h * W + w0 + n];
            float b1 = s[(b * D + (ok1 ? r1 : 0)) * SLICE + h * W + w0 + n];
            b0 = ok0 ? b0 : 0.0f;
            b1 = ok1 ? b1 : 0.0f;
            v2f A = mk2(bandA(j0, n), bandA(j0 + 1, n));
            a = wmma4(A, mk2(b0, b1), a);
        }
        acc[f] = a;
    }

    float lsum = 0.0f;
#pragma unroll
    for (int g = 0; g < 8; ++g) {
        float uI = acc[0][g] * INV_WIN3;
        float uJ = acc[1][g] * INV_WIN3;
        float I2 = acc[2][g] * INV_WIN3 - uI * uI;
        float J2 = acc[3][g] * INV_WIN3 - uJ * uJ;
        float IJ = acc[4][g] * INV_WIN3 - uI * uJ;
        lsum += (IJ * IJ) / (I2 * J2 + EPS);
    }

    red[threadIdx.x] = lsum;
    __syncthreads();
#pragma unroll
    for (int off = 128; off > 0; off >>= 1) {
        if (threadIdx.x < off) red[threadIdx.x] += red[threadIdx.x + off];
        __syncthreads();
    }
    if (threadIdx.x == 0) partial[blockIdx.x] = red[0];
}

// ---------------------------------------------------------------------------
// Final: deterministic sum of per-block partials -> -mean(ncc).
// ---------------------------------------------------------------------------
__global__ __launch_bounds__(256) void ncc_final(const float* __restrict__ partial,
                                                 float* __restrict__ out) {
    __shared__ float red[256];
    float s = 0.0f;
    for (int i = threadIdx.x; i < BLOCKS; i += 256) s += partial[i];
    red[threadIdx.x] = s;
    __syncthreads();
#pragma unroll
    for (int off = 128; off > 0; off >>= 1) {
        if (threadIdx.x < off) red[threadIdx.x] += red[threadIdx.x + off];
        __syncthreads();
    }
    if (threadIdx.x == 0) out[0] = -red[0] / (float)NVOX;
}

// ---------------------------------------------------------------------------
// Workspace layout (floats): [0,5N) bufA | [5N,10N) bufB | [10N,10N+6912) partials
// Requires ws_size >= (10*NVOX + BLOCKS) * 4 bytes (~567 MB).
// ---------------------------------------------------------------------------
extern "C" void kernel_launch(void* const* d_in, const int* in_sizes, int n_in,
                              void* d_out, int out_size, void* d_ws, size_t ws_size,
                              hipStream_t stream) {
    (void)in_sizes; (void)n_in; (void)out_size; (void)ws_size;
    const float* yp = (const float*)d_in[0];
    const float* yt = (const float*)d_in[1];
    float* ws      = (float*)d_ws;
    float* bufA    = ws;
    float* bufB    = ws + (size_t)5 * NVOX;
    float* partial = ws + (size_t)10 * NVOX;

    ncc_pass1<<<BLOCKS, 256, 0, stream>>>(yp, yt, bufA);
    ncc_pass2<<<BLOCKS, 256, 0, stream>>>(bufA, bufB);
    ncc_pass3<<<BLOCKS, 256, 0, stream>>>(bufB, partial);
    ncc_final<<<1, 256, 0, stream>>>(partial, (float*)d_out);
}